// PointTransformerLayerGlobal_84052509982726
// MI455X (gfx1250) — compile-verified
//
#include <hip/hip_runtime.h>
#include <hip/hip_bf16.h>

#define EPSB 1e-5f

typedef float v2f __attribute__((ext_vector_type(2)));
typedef float v8f __attribute__((ext_vector_type(8)));

static __device__ __forceinline__ v8f wmma4(v2f a, v2f b, v8f c) {
  // D = A(16x4,f32) * B(4x16,f32) + C(16x16,f32)
  return __builtin_amdgcn_wmma_f32_16x16x4_f32(false, a, false, b, (short)0, c,
                                               false, false);
}

// -------------------------------------------------------------------------
// Kernel 1: QKV projections  x(2048x64) @ W(64x64) + b  -> ws
// One wave per 16x16 output tile. 3 mats * 128 rowtiles * 4 coltiles = 1536.
// -------------------------------------------------------------------------
__global__ void __launch_bounds__(256) k1_qkv(
    const float* __restrict__ x,
    const float* __restrict__ wq, const float* __restrict__ bq,
    const float* __restrict__ wk, const float* __restrict__ bk,
    const float* __restrict__ wv, const float* __restrict__ bv,
    float* __restrict__ xq, float* __restrict__ xk, float* __restrict__ xv)
{
  const int wave = threadIdx.x >> 5, lane = threadIdx.x & 31;
  const int hi = lane >> 4, lr = lane & 15;
  const int task = blockIdx.x * 8 + wave;        // 0..1535
  const int mat = task >> 9;                     // 0..2
  const int rem = task & 511;
  const int rt = rem >> 2, ct = rem & 3;
  const float* W = (mat == 0) ? wq : (mat == 1) ? wk : wv;
  const float* B = (mat == 0) ? bq : (mat == 1) ? bk : bv;
  float*       O = (mat == 0) ? xq : (mat == 1) ? xk : xv;
  const int r0 = rt * 16;
  const int col = ct * 16 + lr;
  const float* xrow = x + (r0 + lr) * 64;

  v8f acc = {0.f,0.f,0.f,0.f,0.f,0.f,0.f,0.f};
  for (int ks = 0; ks < 16; ++ks) {
    const int kk = ks * 4 + 2 * hi;
    v2f a; a.x = xrow[kk];          a.y = xrow[kk + 1];
    v2f b; b.x = W[kk * 64 + col];  b.y = W[(kk + 1) * 64 + col];
    acc = wmma4(a, b, acc);
  }
  const float bias = B[col];
#pragma unroll
  for (int v = 0; v < 8; ++v)
    O[(r0 + v + hi * 8) * 64 + col] = acc[v] + bias;
}

// -------------------------------------------------------------------------
// Kernel 2: fused pair pipeline -> attention logits (q, 512, 8) in ws.
// One wave per (query, 16-neighbor chunk): 2048*32 = 65536 wave tasks.
// -------------------------------------------------------------------------
__global__ void __launch_bounds__(256) k2_logits(
    const float* __restrict__ p,
    const float* __restrict__ xq, const float* __restrict__ xk,
    const float* __restrict__ wp1, const float* __restrict__ bp1,
    const float* __restrict__ gp, const float* __restrict__ betap,
    const float* __restrict__ mup, const float* __restrict__ varp,
    const float* __restrict__ wp2, const float* __restrict__ bp2,
    const float* __restrict__ g1, const float* __restrict__ beta1,
    const float* __restrict__ mu1, const float* __restrict__ var1,
    const float* __restrict__ ww1, const float* __restrict__ bw1,
    const float* __restrict__ g2, const float* __restrict__ beta2,
    const float* __restrict__ mu2, const float* __restrict__ var2,
    const float* __restrict__ ww2, const float* __restrict__ bw2,
    float* __restrict__ wlog)
{
  __shared__ float stage[8][16 * 64];           // 4KB per wave staging
  const int wave = threadIdx.x >> 5, lane = threadIdx.x & 31;
  const int hi = lane >> 4, lr = lane & 15;
  const int task = blockIdx.x * 8 + wave;       // 65536 total
  const int q = task >> 5;
  const int c = task & 31;
  const int jb = (q >> 9) << 9;                 // batch base row
  const int j0 = jb + (c << 4);                 // chunk base row
  const int jn = j0 + lr;                       // this lane's neighbor
  float* st = stage[wave];

  // --- positional MLP: t = relu(bn(p_r @ wp1 + bp1)) -------------------
  const float m00=wp1[0],m01=wp1[1],m02=wp1[2];
  const float m10=wp1[3],m11=wp1[4],m12=wp1[5];
  const float m20=wp1[6],m21=wp1[7],m22=wp1[8];
  const float s0=gp[0]*rsqrtf(varp[0]+EPSB);
  const float s1=gp[1]*rsqrtf(varp[1]+EPSB);
  const float s2=gp[2]*rsqrtf(varp[2]+EPSB);
  const float h0=betap[0]+(bp1[0]-mup[0])*s0;
  const float h1=betap[1]+(bp1[1]-mup[1])*s1;
  const float h2=betap[2]+(bp1[2]-mup[2])*s2;
  const float pq0=p[q*3], pq1=p[q*3+1], pq2=p[q*3+2];
  const float r0=p[jn*3]-pq0, r1=p[jn*3+1]-pq1, r2=p[jn*3+2]-pq2;
  const float t0=fmaxf((r0*m00+r1*m10+r2*m20)*s0+h0,0.f);
  const float t1=fmaxf((r0*m01+r1*m11+r2*m21)*s1+h1,0.f);
  const float t2=fmaxf((r0*m02+r1*m12+r2*m22)*s2+h2,0.f);

  const v8f vz = {0.f,0.f,0.f,0.f,0.f,0.f,0.f,0.f};
  v2f a; a.x = hi ? t2 : t0; a.y = hi ? 0.f : t1;   // K=3 padded to 4
  v8f pe[4];
#pragma unroll
  for (int g = 0; g < 4; ++g) {
    const int col = g*16 + lr;
    v2f b; b.x = hi ? wp2[2*64+col] : wp2[col];
    b.y = hi ? 0.f : wp2[64+col];
    pe[g] = wmma4(a, b, vz);                        // t @ wp2 (no bp2 yet)
  }

  // --- w_pre = gk - x_q + p_e + bp2 ; bn1 ; relu -> LDS ----------------
#pragma unroll
  for (int g = 0; g < 4; ++g) {
    const int col = g*16 + lr;
    const float sc = g1[col]*rsqrtf(var1[col]+EPSB);
    const float sh = beta1[col]-mu1[col]*sc;
    const float base = bp2[col] - xq[q*64+col];
#pragma unroll
    for (int v = 0; v < 8; ++v) {
      const int row = v + hi*8;
      const float wv = xk[(j0+row)*64+col] + base + pe[g][v];
      st[row*64+col] = fmaxf(wv*sc + sh, 0.f);
    }
  }
  __syncthreads();

  // --- GEMM1: (16x64) @ ww1(64x8 padded to 16) -------------------------
  v8f w1 = vz;
  for (int ks = 0; ks < 16; ++ks) {
    const int kk = ks*4 + 2*hi;
    v2f a1; a1.x = st[lr*64+kk]; a1.y = st[lr*64+kk+1];
    v2f b1; b1.x = 0.f; b1.y = 0.f;
    if (lr < 8) { b1.x = ww1[kk*8+lr]; b1.y = ww1[(kk+1)*8+lr]; }
    w1 = wmma4(a1, b1, w1);
  }
  __syncthreads();

  // --- +bw1 ; bn2 ; relu -> LDS (16x8) ---------------------------------
  if (lr < 8) {
    const float sc2 = g2[lr]*rsqrtf(var2[lr]+EPSB);
    const float sh2 = beta2[lr]-mu2[lr]*sc2;
    const float bb  = bw1[lr];
#pragma unroll
    for (int v = 0; v < 8; ++v) {
      const int row = v + hi*8;
      st[row*8+lr] = fmaxf((w1[v]+bb)*sc2 + sh2, 0.f);
    }
  }
  __syncthreads();

  // --- GEMM2: (16x8) @ ww2(8x8 padded) ---------------------------------
  v8f w2 = vz;
#pragma unroll
  for (int ks = 0; ks < 2; ++ks) {
    const int kk = ks*4 + 2*hi;
    v2f a2; a2.x = st[lr*8+kk]; a2.y = st[lr*8+kk+1];
    v2f b2; b2.x = 0.f; b2.y = 0.f;
    if (lr < 8) { b2.x = ww2[kk*8+lr]; b2.y = ww2[(kk+1)*8+lr]; }
    w2 = wmma4(a2, b2, w2);
  }

  if (lr < 8) {
    const float bb2 = bw2[lr];
#pragma unroll
    for (int v = 0; v < 8; ++v) {
      const int row = v + hi*8;
      wlog[((size_t)q*512 + (c<<4) + row)*8 + lr] = w2[v] + bb2;
    }
  }
}

// -------------------------------------------------------------------------
// Kernel 3: softmax stats over neighbor dim: max + 1/sumexp per (q, cw)
// -------------------------------------------------------------------------
__global__ void __launch_bounds__(256) k3_softmax(
    const float* __restrict__ wlog, float* __restrict__ mx, float* __restrict__ sm)
{
  const int t = blockIdx.x * 256 + threadIdx.x;   // 16384 = 2048*8
  const int q = t >> 3, cw = t & 7;
  const float* base = wlog + (size_t)q*512*8 + cw;
  float m = -1e30f;
  for (int j = 0; j < 512; ++j) m = fmaxf(m, base[j*8]);
  float s = 0.f;
  for (int j = 0; j < 512; ++j) s += __expf(base[j*8] - m);
  mx[t] = m;
  sm[t] = 1.f / s;
}

// -------------------------------------------------------------------------
// Kernel 4: weighted sum out[q, s*8+cw] = sum_j softmax(w)[j,cw]*(gv+pe)
// One wave per query (pe recomputed per chunk with WMMA).
// -------------------------------------------------------------------------
__global__ void __launch_bounds__(256) k4_out(
    const float* __restrict__ p, const float* __restrict__ xv,
    const float* __restrict__ wp1, const float* __restrict__ bp1,
    const float* __restrict__ gp, const float* __restrict__ betap,
    const float* __restrict__ mup, const float* __restrict__ varp,
    const float* __restrict__ wp2, const float* __restrict__ bp2,
    const float* __restrict__ wlog, const float* __restrict__ mx,
    const float* __restrict__ sm, float* __restrict__ out)
{
  const int wave = threadIdx.x >> 5, lane = threadIdx.x & 31;
  const int hi = lane >> 4, lr = lane & 15;
  const int q = blockIdx.x * 8 + wave;            // 2048
  const int jb = (q >> 9) << 9;
  const int cwl = lr & 7;
  const float mm  = mx[q*8+cwl];
  const float inv = sm[q*8+cwl];

  const float m00=wp1[0],m01=wp1[1],m02=wp1[2];
  const float m10=wp1[3],m11=wp1[4],m12=wp1[5];
  const float m20=wp1[6],m21=wp1[7],m22=wp1[8];
  const float s0=gp[0]*rsqrtf(varp[0]+EPSB);
  const float s1=gp[1]*rsqrtf(varp[1]+EPSB);
  const float s2=gp[2]*rsqrtf(varp[2]+EPSB);
  const float h0=betap[0]+(bp1[0]-mup[0])*s0;
  const float h1=betap[1]+(bp1[1]-mup[1])*s1;
  const float h2=betap[2]+(bp1[2]-mup[2])*s2;
  const float pq0=p[q*3], pq1=p[q*3+1], pq2=p[q*3+2];

  float bpc[4];
#pragma unroll
  for (int g = 0; g < 4; ++g) bpc[g] = bp2[g*16+lr];

  const v8f vz = {0.f,0.f,0.f,0.f,0.f,0.f,0.f,0.f};
  float acc[4] = {0.f, 0.f, 0.f, 0.f};

  for (int c = 0; c < 32; ++c) {
    const int j0 = jb + c*16;
    const int jn = j0 + lr;
    if (c < 31) __builtin_prefetch(&xv[(j0+16)*64], 0, 1);  // global_prefetch_b8

    const float r0=p[jn*3]-pq0, r1=p[jn*3+1]-pq1, r2=p[jn*3+2]-pq2;
    const float t0=fmaxf((r0*m00+r1*m10+r2*m20)*s0+h0,0.f);
    const float t1=fmaxf((r0*m01+r1*m11+r2*m21)*s1+h1,0.f);
    const float t2=fmaxf((r0*m02+r1*m12+r2*m22)*s2+h2,0.f);
    v2f a; a.x = hi ? t2 : t0; a.y = hi ? 0.f : t1;

    v8f pe[4];
#pragma unroll
    for (int g = 0; g < 4; ++g) {
      const int col = g*16 + lr;
      v2f b; b.x = hi ? wp2[2*64+col] : wp2[col];
      b.y = hi ? 0.f : wp2[64+col];
      pe[g] = wmma4(a, b, vz);
    }

    float e[8];
#pragma unroll
    for (int v = 0; v < 8; ++v) {
      const int row = v + hi*8;
      e[v] = __expf(wlog[((size_t)q*512 + c*16 + row)*8 + cwl] - mm) * inv;
    }

#pragma unroll
    for (int g = 0; g < 4; ++g) {
      const int col = g*16 + lr;
#pragma unroll
      for (int v = 0; v < 8; ++v) {
        const int row = v + hi*8;
        acc[g] += e[v] * (xv[(j0+row)*64+col] + bpc[g] + pe[g][v]);
      }
    }
  }

#pragma unroll
  for (int g = 0; g < 4; ++g) acc[g] += __shfl_xor(acc[g], 16, 32);
  if (lane < 16) {
#pragma unroll
    for (int g = 0; g < 4; ++g) out[q*64 + g*16 + lane] = acc[g];
  }
}

// -------------------------------------------------------------------------
extern "C" void kernel_launch(void* const* d_in, const int* in_sizes, int n_in,
                              void* d_out, int out_size, void* d_ws, size_t ws_size,
                              hipStream_t stream)
{
  const float* p    = (const float*)d_in[0];
  const float* x    = (const float*)d_in[1];
  // d_in[2] = nsample (fixed 512 in this problem)
  const float* wq   = (const float*)d_in[3];  const float* bq    = (const float*)d_in[4];
  const float* wk   = (const float*)d_in[5];  const float* bk    = (const float*)d_in[6];
  const float* wv   = (const float*)d_in[7];  const float* bv    = (const float*)d_in[8];
  const float* wp1  = (const float*)d_in[9];  const float* bp1   = (const float*)d_in[10];
  const float* gp   = (const float*)d_in[11]; const float* betap = (const float*)d_in[12];
  const float* mup  = (const float*)d_in[13]; const float* varp  = (const float*)d_in[14];
  const float* wp2  = (const float*)d_in[15]; const float* bp2   = (const float*)d_in[16];
  const float* g1   = (const float*)d_in[17]; const float* beta1 = (const float*)d_in[18];
  const float* mu1  = (const float*)d_in[19]; const float* var1  = (const float*)d_in[20];
  const float* ww1  = (const float*)d_in[21]; const float* bw1   = (const float*)d_in[22];
  const float* g2   = (const float*)d_in[23]; const float* beta2 = (const float*)d_in[24];
  const float* mu2  = (const float*)d_in[25]; const float* var2  = (const float*)d_in[26];
  const float* ww2  = (const float*)d_in[27]; const float* bw2   = (const float*)d_in[28];

  float* ws   = (float*)d_ws;
  float* xq   = ws;                       // 2048*64
  float* xk   = ws + 131072;              // 2048*64
  float* xvp  = ws + 262144;              // 2048*64
  float* wlog = ws + 393216;              // 2048*512*8
  float* mxp  = ws + 393216 + 8388608;    // 2048*8
  float* smp  = mxp + 16384;              // 2048*8
  float* out  = (float*)d_out;

  k1_qkv<<<192, 256, 0, stream>>>(x, wq, bq, wk, bk, wv, bv, xq, xk, xvp);
  k2_logits<<<8192, 256, 0, stream>>>(p, xq, xk, wp1, bp1, gp, betap, mup, varp,
                                      wp2, bp2, g1, beta1, mu1, var1,
                                      ww1, bw1, g2, beta2, mu2, var2, ww2, bw2,
                                      wlog);
  k3_softmax<<<64, 256, 0, stream>>>(wlog, mxp, smp);
  k4_out<<<256, 256, 0, stream>>>(p, xvp, wp1, bp1, gp, betap, mup, varp,
                                  wp2, bp2, wlog, mxp, smp, out);
}